// MaxUnpooling2D_53214644797867
// MI455X (gfx1250) — compile-verified
//
#include <hip/hip_runtime.h>

// MaxUnpooling2D scatter-add for fixed shapes:
//   updates [16,128,128,64] f32, mask [16,128,128,64] i32 (flat index into [256,256,64])
//   out     [16,256,256,64] f32  = zeros.at[b, m>>14, (m>>6)&255, c].add(updates)
// Power-of-two dims:
//   H*W*C      = 2^20  (per-batch input elements)
//   Ho*Wo*C    = 2^22  (per-batch output elements)
//   (y*Wo+x)*C = mask & ~63
typedef __attribute__((ext_vector_type(4))) float v4f;
typedef __attribute__((ext_vector_type(4))) int   v4i;

#define C_MASK         63u         // C-1
#define IN_PER_B_LOG2  20u         // log2(128*128*64)
#define OUT_PER_B_LOG2 22u         // log2(256*256*64)

// ---------------------------------------------------------------------------
// Scatter pass (placed FIRST so the disasm snippet shows the atomic lowering):
// stream updates+mask with 128-bit NT loads (4 consecutive channels/thread),
// decode addr = b*2^22 + (mask & ~63) + channel, issue relaxed agent-scope
// f32 atomic adds with discarded result (-> no-return global_atomic_add_f32).
// ---------------------------------------------------------------------------
__global__ __launch_bounds__(256) void unpool_scatter_kernel(
    const float* __restrict__ updates, const int* __restrict__ mask,
    float* __restrict__ out, int ngroups) {
  int g = blockIdx.x * blockDim.x + threadIdx.x;
  if (g >= ngroups) return;

  v4f u = __builtin_nontemporal_load(reinterpret_cast<const v4f*>(updates) + g);
  v4i m = __builtin_nontemporal_load(reinterpret_cast<const v4i*>(mask) + g);

  const unsigned base_elem = (unsigned)g << 2;                    // flat input idx
  const unsigned b         = base_elem >> IN_PER_B_LOG2;          // batch
  const unsigned f0        = base_elem & C_MASK;                  // first channel
  const unsigned bofs      = b << OUT_PER_B_LOG2;                 // batch offset in out

#pragma unroll
  for (int j = 0; j < 4; ++j) {
    const unsigned mm   = (unsigned)m[j];
    const unsigned addr = bofs + (mm & ~C_MASK) + (f0 + (unsigned)j);
    (void)__hip_atomic_fetch_add(out + addr, u[j],
                                 __ATOMIC_RELAXED, __HIP_MEMORY_SCOPE_AGENT);
  }
}

// ---------------------------------------------------------------------------
// Zero pass: 128-bit non-temporal stores (268 MB stream; don't thrash L2 with
// lines we won't read back before the scatter pass).
// ---------------------------------------------------------------------------
__global__ __launch_bounds__(256) void unpool_zero_kernel(float* __restrict__ out,
                                                          int n4, int n) {
  const int stride = gridDim.x * blockDim.x;
  v4f z = (v4f)0.0f;
  v4f* __restrict__ out4 = reinterpret_cast<v4f*>(out);
  for (int i = blockIdx.x * blockDim.x + threadIdx.x; i < n4; i += stride) {
    __builtin_nontemporal_store(z, out4 + i);
  }
  // scalar tail (n is a multiple of 4 for the reference shapes; kept for safety)
  if (blockIdx.x == 0 && threadIdx.x == 0) {
    for (int i = n4 * 4; i < n; ++i) out[i] = 0.0f;
  }
}

// ---------------------------------------------------------------------------
extern "C" void kernel_launch(void* const* d_in, const int* in_sizes, int n_in,
                              void* d_out, int out_size, void* d_ws, size_t ws_size,
                              hipStream_t stream) {
  const float* updates = (const float*)d_in[0];
  const int*   mask    = (const int*)d_in[1];
  float*       out     = (float*)d_out;

  // zero-fill: grid-strided, 8 x 128-bit stores per thread
  const int n4 = out_size >> 2;
  const int zthreads = 256;
  int zblocks = (n4 + zthreads * 8 - 1) / (zthreads * 8);
  if (zblocks < 1) zblocks = 1;
  unpool_zero_kernel<<<zblocks, zthreads, 0, stream>>>(out, n4, out_size);

  // scatter: one thread per 4 consecutive channels
  const int ngroups = in_sizes[0] / 4;
  const int sthreads = 256;
  const int sblocks = (ngroups + sthreads - 1) / sthreads;
  unpool_scatter_kernel<<<sblocks, sthreads, 0, stream>>>(updates, mask, out, ngroups);
}